// LogSpaceSinkhorn_3848290697877
// MI455X (gfx1250) — compile-verified
//
#include <hip/hip_runtime.h>

// Log-space Sinkhorn, fused via dual potentials u,v (+w for symmetric final).
// 64 matrices of 1024x1024 f32. 13 read passes + 1 write pass over the tensor.
// Passes alternate traversal direction (row fwd / col rev) so each pass starts
// on data still resident in the 192MB L2 (tensor is 256MB): ~75% L2 reuse.

#define MATS 64
#define N 1024

typedef float v4f __attribute__((ext_vector_type(4)));
typedef float v2f __attribute__((ext_vector_type(2)));
typedef int   v4i __attribute__((ext_vector_type(4)));
typedef __attribute__((address_space(1))) v4i g_v4i;  // global int4
typedef __attribute__((address_space(3))) v4i l_v4i;  // LDS int4

#if defined(__has_builtin)
#if __has_builtin(__builtin_amdgcn_global_load_async_to_lds_b128) && \
    __has_builtin(__builtin_amdgcn_s_wait_asynccnt)
#define HAVE_ASYNC_LDS 1
#endif
#endif
#ifndef HAVE_ASYNC_LDS
#define HAVE_ASYNC_LDS 0
#endif

// ---- online log-sum-exp helpers (branch-free, -inf safe) -------------------
__device__ __forceinline__ void lse_acc(float x, float& m, float& s) {
  float M = fmaxf(m, x);
  s = s * __expf(m - M) + __expf(x - M);
  m = M;
}

__device__ __forceinline__ void lse_merge(float om, float os, float& m, float& s) {
  float M = fmaxf(m, om);
  s = s * __expf(m - M) + os * __expf(om - M);
  m = M;
}

__device__ __forceinline__ void lse_wave_reduce(float& m, float& s) {
#pragma unroll
  for (int off = 16; off > 0; off >>= 1) {
    float om = __shfl_xor(m, off, 32);
    float os = __shfl_xor(s, off, 32);
    lse_merge(om, os, m, s);
  }
}

// Stage 1024 floats (one potential vector) into LDS with 256 threads,
// using gfx1250 async global->LDS DMA when available.
__device__ __forceinline__ void stage_pot(const float* __restrict__ src,
                                          v4f* lds, int tid) {
  const v4f* s4 = reinterpret_cast<const v4f*>(src);
#if HAVE_ASYNC_LDS
  __builtin_amdgcn_global_load_async_to_lds_b128(
      (g_v4i*)(v4f*)(s4 + tid),
      (l_v4i*)(v4f*)(lds + tid),
      /*offset=*/0, /*cpol=*/0);
  __builtin_amdgcn_s_wait_asynccnt(0);
#else
  lds[tid] = s4[tid];
#endif
  __syncthreads();
}

// ---- kernels ---------------------------------------------------------------
__global__ void k_init(float* __restrict__ p, int n) {
  int i = blockIdx.x * blockDim.x + threadIdx.x;
  if (i < n) p[i] = 0.0f;
}

// Row pass: upot[m][i] = -LSE_j( K[m][i][j]*inv + vpot[m][j] )
// grid = MATS*32 blocks, 256 threads; block covers 32 rows of one matrix.
// Traverses matrices FORWARD (0..63).
__global__ void k_row(const float* __restrict__ K, const float* __restrict__ tau,
                      const float* __restrict__ vpot, float* __restrict__ upot) {
  __shared__ v4f vl[256];
  const int mtx = blockIdx.x >> 5;
  const int rb  = blockIdx.x & 31;
  const int tid = threadIdx.x;

  stage_pot(vpot + mtx * N, vl, tid);

  const float inv  = 1.0f / fmaxf(tau[0], 0.1f);
  const int wave = tid >> 5, lane = tid & 31;
  const float* Kb = K + ((size_t)mtx << 20);

#pragma unroll
  for (int r = 0; r < 4; ++r) {
    const int row = rb * 32 + wave * 4 + r;
    const v4f* Kr = reinterpret_cast<const v4f*>(Kb + (size_t)row * N);
    float m = -__builtin_inff(), s = 0.0f;
#pragma unroll
    for (int t = 0; t < 8; ++t) {
      const int q = t * 32 + lane;        // fully coalesced b128 per wave
      v4f k4 = Kr[q];
      v4f v4 = vl[q];
      lse_acc(__builtin_fmaf(k4.x, inv, v4.x), m, s);
      lse_acc(__builtin_fmaf(k4.y, inv, v4.y), m, s);
      lse_acc(__builtin_fmaf(k4.z, inv, v4.z), m, s);
      lse_acc(__builtin_fmaf(k4.w, inv, v4.w), m, s);
    }
    lse_wave_reduce(m, s);
    if (lane == 0) upot[mtx * N + row] = -(m + __logf(s));
  }
}

// Col pass: outpot[m][j] = -LSE_i( K[m][i][j]*inv + uin[m][i] )
// grid = MATS*16 blocks, 256 threads; block covers a 64-column strip
// (float2 per lane), 8 waves split the 1024 rows (128 each), partials
// merged through LDS. Traverses matrices in REVERSE (63..0) for L2 reuse.
__global__ void k_col(const float* __restrict__ K, const float* __restrict__ tau,
                      const float* __restrict__ uin, float* __restrict__ outpot) {
  __shared__ float ul[N];
  __shared__ float mP[8][64];
  __shared__ float sP[8][64];
  const int mtx   = (MATS - 1) - (blockIdx.x >> 4);   // reverse traversal
  const int strip = blockIdx.x & 15;
  const int tid   = threadIdx.x;

  stage_pot(uin + mtx * N, reinterpret_cast<v4f*>(ul), tid);

  const float inv = 1.0f / fmaxf(tau[0], 0.1f);
  const int wave = tid >> 5, lane = tid & 31;
  const int col0 = strip * 64 + lane * 2;             // coalesced 256B per wave
  const float* Kc = K + ((size_t)mtx << 20) + col0;

  float m0 = -__builtin_inff(), s0 = 0.0f;
  float m1 = -__builtin_inff(), s1 = 0.0f;
  const int i0 = wave * 128;
  for (int i = i0; i < i0 + 128; i += 4) {
    __builtin_prefetch(Kc + (size_t)(i + 16) * N, 0, 3);  // 64KB ahead
#pragma unroll
    for (int j = 0; j < 4; ++j) {
      v2f k2 = *reinterpret_cast<const v2f*>(Kc + (size_t)(i + j) * N);
      float uu = ul[i + j];
      lse_acc(__builtin_fmaf(k2.x, inv, uu), m0, s0);
      lse_acc(__builtin_fmaf(k2.y, inv, uu), m1, s1);
    }
  }
  mP[wave][lane * 2]     = m0;
  sP[wave][lane * 2]     = s0;
  mP[wave][lane * 2 + 1] = m1;
  sP[wave][lane * 2 + 1] = s1;
  __syncthreads();
  if (tid < 64) {
    float M = mP[0][tid], S = sP[0][tid];
#pragma unroll
    for (int w2 = 1; w2 < 8; ++w2) lse_merge(mP[w2][tid], sP[w2][tid], M, S);
    outpot[mtx * N + strip * 64 + tid] = -(M + __logf(S));
  }
}

// Final: out = exp( K*inv + u_i + 0.5*(v_j + w_j) ), nontemporal store.
// Forward traversal (matrix 0 first) — matches end state of preceding col pass.
__global__ void k_fin(const float* __restrict__ K, const float* __restrict__ tau,
                      const float* __restrict__ u, const float* __restrict__ v,
                      const float* __restrict__ w, float* __restrict__ out) {
  const long long gid  = (long long)blockIdx.x * blockDim.x + threadIdx.x;
  const long long base = gid * 4;
  const int mtx  = (int)(base >> 20);
  const int rem  = (int)(base & ((1 << 20) - 1));
  const int row  = rem >> 10;
  const int colj = rem & (N - 1);

  const float inv = 1.0f / fmaxf(tau[0], 0.1f);
  v4f k4 = *reinterpret_cast<const v4f*>(K + base);
  const float uu = u[mtx * N + row];
  v4f v4 = *reinterpret_cast<const v4f*>(v + mtx * N + colj);
  v4f w4 = *reinterpret_cast<const v4f*>(w + mtx * N + colj);

  v4f o;
  o.x = __expf(__builtin_fmaf(k4.x, inv, uu) + 0.5f * (v4.x + w4.x));
  o.y = __expf(__builtin_fmaf(k4.y, inv, uu) + 0.5f * (v4.y + w4.y));
  o.z = __expf(__builtin_fmaf(k4.z, inv, uu) + 0.5f * (v4.z + w4.z));
  o.w = __expf(__builtin_fmaf(k4.w, inv, uu) + 0.5f * (v4.w + w4.w));
  __builtin_nontemporal_store(o, reinterpret_cast<v4f*>(out + base));
}

// ---- launch ----------------------------------------------------------------
extern "C" void kernel_launch(void* const* d_in, const int* in_sizes, int n_in,
                              void* d_out, int out_size, void* d_ws, size_t ws_size,
                              hipStream_t stream) {
  const float* K   = (const float*)d_in[0];   // (4,16,1024,1024) f32 -> 64 matrices
  const float* tau = (const float*)d_in[1];   // scalar on device
  float* out = (float*)d_out;

  float* u = (float*)d_ws;                    // [64][1024]
  float* v = u + MATS * N;                    // [64][1024]
  float* w = v + MATS * N;                    // [64][1024]   (768 KB total)

  const int npot = 3 * MATS * N;
  k_init<<<(npot + 255) / 256, 256, 0, stream>>>(u, npot);

  for (int it = 0; it < 5; ++it) {
    k_row<<<MATS * 32, 256, 0, stream>>>(K, tau, v, u);  // u <- rowpass(v)   fwd
    k_col<<<MATS * 16, 256, 0, stream>>>(K, tau, u, v);  // v <- colpass(u)   rev
  }
  // symmetric final: u' <- rowpass(v); w <- colpass(u'); out = exp(K+u'+(v+w)/2)
  k_row<<<MATS * 32, 256, 0, stream>>>(K, tau, v, u);
  k_col<<<MATS * 16, 256, 0, stream>>>(K, tau, u, w);

  const long long total = (long long)MATS * N * N;  // 64M elements
  k_fin<<<(unsigned)(total / 4 / 256), 256, 0, stream>>>(K, tau, u, v, w, out);
}